// EquivariantSubsample_53317724013297
// MI455X (gfx1250) — compile-verified
//
#include <hip/hip_runtime.h>

// EquivariantSubsample: per-(b,c) shifted 2x2 max pool.
//   images: [16,64,512,512] f32, p_w/p_h: [16,64] i32 in {0,1}
//   out:    [16,64,256,256] f32
// Memory-bound: 1.25 GiB traffic -> ~57us floor at 23.3 TB/s.
// Data path: TDM (tensor_load_to_lds) stages input row-strips into LDS,
// compute does the shifted 2x2 max from LDS, non-temporal stores stream out.

#define BB 16
#define CC 64
#define HH 512
#define WW 512
#define HRD 2
#define WRD 2
#define OUT_H (HH / HRD)            // 256
#define OUT_W (WW / WRD)            // 256
#define ROWS_PER_BLOCK 8            // output rows per workgroup strip
#define IN_ROWS (2 * ROWS_PER_BLOCK + 1)  // 17 input rows staged in LDS
#define STRIPS (OUT_H / ROWS_PER_BLOCK)   // 32
#define NTHREADS 256

typedef unsigned int u32x4 __attribute__((ext_vector_type(4)));
typedef int i32x8 __attribute__((ext_vector_type(8)));
typedef int i32x4 __attribute__((ext_vector_type(4)));

__global__ __launch_bounds__(NTHREADS)
void EquivariantSubsample_pool_kernel(const float* __restrict__ images,
                                      const int* __restrict__ p_w,
                                      const int* __restrict__ p_h,
                                      float* __restrict__ out) {
    __shared__ float tile[IN_ROWS * WW];   // 17 * 512 * 4B = 34 KB

    const int plane  = (int)blockIdx.x >> 5;        // 0..1023  (b*c)
    const int strip  = (int)blockIdx.x & (STRIPS - 1);
    const int oy0    = strip * ROWS_PER_BLOCK;
    const int row_lo = oy0 * HRD;                   // first staged input row

    // Block-uniform sub-pixel offsets -> scalar loads.
    const int pw = p_w[plane];
    const int ph = p_h[plane];

    const unsigned long long gaddr =
        (unsigned long long)(const void*)(images + ((size_t)plane * HH + row_lo) * (size_t)WW);
    const int rows_left = HH - row_lo;   // TDM OOB rows (y >= rows_left) read as zero

#if defined(__gfx1250__) && __has_builtin(__builtin_amdgcn_tensor_load_to_lds)
    if (threadIdx.x == 0) {
        // Tensor DMA descriptor (ISA 8.3-8.5), 2-D tile of f32:
        //   tile_dim0 = 512 cols, tile_dim1 = IN_ROWS rows, stride = 512.
        const unsigned lds_off = (unsigned)(unsigned long long)(const void*)&tile[0];
        u32x4 g0 = {
            0x1u,                                   // count=1, load, user mode
            lds_off,                                // lds_addr (bytes)
            (unsigned)(gaddr & 0xffffffffu),        // global_addr[31:0]
            (unsigned)((gaddr >> 32) & 0x1ffffffu)  // global_addr[56:32]
                | (2u << 30)                        // type = 2 (image)
        };
        const unsigned long long d1s = (unsigned long long)HH * WW; // dim1 stride (elems)
        i32x8 g1 = {
            (int)(2u << 16),                              // wg_mask=0, data_size=2 (4B)
            (int)((WW & 0xffffu) << 16),                  // barrier=0 | tensor_dim0.lo16
            (int)((WW >> 16) | ((rows_left & 0xffff) << 16)),   // dim0.hi16 | tensor_dim1.lo16
            (int)(((unsigned)rows_left >> 16) | ((WW & 0xffffu) << 16)), // dim1.hi16 | tile_dim0
            (int)IN_ROWS,                                 // tile_dim1 | tile_dim2=0
            (int)WW,                                      // tensor_dim0_stride[31:0]
            (int)(((d1s & 0xffffu) << 16) | 0u),          // dim0_stride[47:32]=0 | dim1_stride.lo16
            (int)(d1s >> 16)                              // tensor_dim1_stride[47:16]
        };
        i32x4 g2 = {0, 0, 0, 0};
        i32x4 g3 = {0, 0, 0, 0};
        i32x8 g4 = {0, 0, 0, 0, 0, 0, 0, 0};   // VADDR4 slot: unused per ISA (NULL)
        __builtin_amdgcn_tensor_load_to_lds(g0, g1, g2, g3, g4, 0);
        __builtin_amdgcn_s_wait_tensorcnt(0);
    }
#else
    // Fallback: the staged rows are contiguous in memory -> flat coalesced copy.
    {
        const float* src = images + ((size_t)plane * HH + row_lo) * (size_t)WW;
        const int n = (rows_left < IN_ROWS ? rows_left : IN_ROWS) * WW;
        for (int i = (int)threadIdx.x; i < n; i += NTHREADS)
            tile[i] = __builtin_nontemporal_load(&src[i]);
    }
#endif
    __syncthreads();

    // One output column per lane: consecutive lanes -> consecutive ox
    // (coalesced stores; stride-2 dword LDS reads are bank-conflict-free).
    const int ox = (int)threadIdx.x;                  // 0..255 == OUT_W
    int x0 = 2 * ox + pw;
    if (x0 > WW - WRD) x0 = WW - WRD;                 // clamp (last column only, ox=255 with pw=1)

    const size_t out_base = (size_t)plane * (OUT_H * OUT_W) + (size_t)oy0 * OUT_W + ox;

#pragma unroll
    for (int r = 0; r < ROWS_PER_BLOCK; ++r) {
        const int oy = oy0 + r;
        int y0 = 2 * oy + ph;
        if (y0 > HH - HRD) y0 = HH - HRD;             // clamp (last block only)
        const int yl = y0 - row_lo;                   // local row in LDS tile
        const float* r0 = &tile[yl * WW + x0];
        const float* r1 = r0 + WW;
        const float m = fmaxf(fmaxf(r0[0], r0[1]), fmaxf(r1[0], r1[1]));
        __builtin_nontemporal_store(m, &out[out_base + (size_t)r * OUT_W]);
    }
}

extern "C" void kernel_launch(void* const* d_in, const int* in_sizes, int n_in,
                              void* d_out, int out_size, void* d_ws, size_t ws_size,
                              hipStream_t stream) {
    (void)in_sizes; (void)n_in; (void)out_size; (void)d_ws; (void)ws_size;
    const float* images = (const float*)d_in[0];
    const int*   p_w    = (const int*)d_in[1];
    const int*   p_h    = (const int*)d_in[2];
    float*       out    = (float*)d_out;

    dim3 grid(BB * CC * STRIPS);   // 1024 planes * 32 strips = 32768 blocks
    dim3 block(NTHREADS);
    EquivariantSubsample_pool_kernel<<<grid, block, 0, stream>>>(images, p_w, p_h, out);
}